// Model_10230612099901
// MI455X (gfx1250) — compile-verified
//
#include <hip/hip_runtime.h>

// out[i] = sum_j S[i,j] * w[j]   (S: 16384x16384 fp32, memory-bound GEMV)
//
// Roofline: 1.07 GB read @ 0.5 FLOP/byte -> pure HBM stream, ~46 us floor at
// 23.3 TB/s. Reduction is done with V_WMMA_F32_16X16X4_F32 (fp32 in / fp32
// accumulate) at negligible compute cost; B is the weight vector broadcast
// across all 16 N-columns so every column of D equals the partial row sum.
//
// K-permutation trick: the WMMA dot product is invariant under any consistent
// (lane-half, vgpr) -> memory-column mapping applied to BOTH A and B, so each
// lane issues one b128 load of 4 contiguous row floats (lanes 16..31 shifted
// +4 columns) feeding two WMMAs -- wide loads, no LDS transpose, and each
// 128B line is fully consumed across 4 load instructions (WGP$ reuse).
//
// Two independent accumulators break the WMMA D->C RAW chain (asm previously
// showed a VALU_DEP_1 stall between every WMMA pair); merged with one v8f add.

#define N        16384
#define WAVES    8            // waves per block
#define KSLICE   (N / WAVES)  // 2048 columns per wave

typedef float v2f __attribute__((ext_vector_type(2)));
typedef float v4f __attribute__((ext_vector_type(4)));
typedef float v8f __attribute__((ext_vector_type(8)));

__global__ __launch_bounds__(32 * WAVES)
void gemv_rowsum_wmma(const float* __restrict__ S,
                      const float* __restrict__ w,
                      float* __restrict__ out) {
    __shared__ float red[WAVES][16];

    const int tid  = threadIdx.x;
    const int lane = tid & 31;
    const int wave = tid >> 5;
    const int m16  = blockIdx.x * 16;          // 16-row tile of this block

    const int mInTile = lane & 15;             // A-matrix M index for this lane
    const int kOff    = (lane >> 4) << 2;      // +4 columns for lanes 16..31

    const float* rowPtr = S + (size_t)(m16 + mInTile) * N;
    const int    kBase  = wave * KSLICE;       // this wave's column slice

    v8f acc0 = {};
    v8f acc1 = {};

    for (int k = kBase; k < kBase + KSLICE; k += 32) {
        // WGP-scope prefetch: fill all cache levels ahead of the stream
        __builtin_prefetch(rowPtr + k + 256, 0, 3);   // global_prefetch_b8
#pragma unroll
        for (int kk = 0; kk < 32; kk += 8) {
            const int c = k + kk + kOff;
            // 16B per lane; lanes L / L+16 cover 32 contiguous bytes per row
            v4f a = *(const v4f*)(rowPtr + c);
            v4f b = *(const v4f*)(w + c);

            v2f a0; a0.x = a.x; a0.y = a.y;
            v2f a1; a1.x = a.z; a1.y = a.w;
            v2f b0; b0.x = b.x; b0.y = b.y;
            v2f b1; b1.x = b.z; b1.y = b.w;

            // D = A*B + C, fp32 16x16x4; two independent accumulator chains
            acc0 = __builtin_amdgcn_wmma_f32_16x16x4_f32(
                       false, a0, false, b0, (short)0, acc0, false, false);
            acc1 = __builtin_amdgcn_wmma_f32_16x16x4_f32(
                       false, a1, false, b1, (short)0, acc1, false, false);
        }
    }

    v8f acc = acc0 + acc1;   // elementwise VALU merge; column redundancy kept

    // All N-columns of D are identical (B broadcast). Per 32-bit C/D layout:
    // lane 0 vgpr v = rowsum(m16+v), lane 16 vgpr v = rowsum(m16+8+v).
    if (lane == 0) {
#pragma unroll
        for (int r = 0; r < 8; ++r) red[wave][r] = acc[r];
    } else if (lane == 16) {
#pragma unroll
        for (int r = 0; r < 8; ++r) red[wave][8 + r] = acc[r];
    }
    __syncthreads();

    if (tid < 16) {
        float s = 0.0f;
#pragma unroll
        for (int wv = 0; wv < WAVES; ++wv) s += red[wv][tid];
        out[m16 + tid] = s;
    }
}

extern "C" void kernel_launch(void* const* d_in, const int* in_sizes, int n_in,
                              void* d_out, int out_size, void* d_ws, size_t ws_size,
                              hipStream_t stream) {
    (void)in_sizes; (void)n_in; (void)out_size; (void)d_ws; (void)ws_size;
    const float* S = (const float*)d_in[0];
    const float* w = (const float*)d_in[1];
    float*     out = (float*)d_out;

    dim3 grid(N / 16);          // 1024 blocks x 8 waves = 8192 streaming waves
    dim3 block(32 * WAVES);     // 256 threads, wave32
    hipLaunchKernelGGL(gemv_rowsum_wmma, grid, block, 0, stream, S, w, out);
}